// DeepSeekMoE_74019466379281
// MI455X (gfx1250) — compile-verified
//
#include <hip/hip_runtime.h>
#include <hip/hip_bf16.h>
#include <math.h>

// ---------------------------------------------------------------------------
// DeepSeek-style MoE (top-2 of 8 routed + 1 shared expert), fp32.
// Memory-bound (~0.3 GB traffic @ 23.3 TB/s => ~13 us floor); fp32 WMMA
// (16x16x4) is far above the bandwidth roofline, so keep fp32 end-to-end.
// LDS layout: A row-major (K-pairs contiguous), B K-pair-packed so every
// WMMA operand is one aligned ds_load_b64 into an even VGPR pair.
// ---------------------------------------------------------------------------

typedef __attribute__((ext_vector_type(2))) float v2f;
typedef __attribute__((ext_vector_type(4))) float v4f;
typedef __attribute__((ext_vector_type(8))) float v8f;

static constexpr int T = 4096;   // B*S tokens
static constexpr int H = 1024;   // hidden
static constexpr int I = 2048;   // intermediate
static constexpr int E = 8;      // routed experts

static constexpr int BM = 64;    // token tile
static constexpr int BN = 64;    // output-feature tile
static constexpr int BK = 16;    // K tile (4 WMMA k-steps)
static constexpr int BP = BK / 2;        // K-pair rows in LDS B tile
static constexpr int LDA = 20;           // As row stride: 80B, 16B aligned
static constexpr int BSTRIDE = 2 * BN + 32;  // 160 floats: +32 banks => the two
                                             // half-wave K-pair groups hit
                                             // disjoint bank halves

// ---------------------------------------------------------------------------
// Router: sigmoid(x @ router_w^T + bias), top-2, renormalize -> dense W[t][8]
// ---------------------------------------------------------------------------
__global__ void moe_router(const float* __restrict__ x,
                           const float* __restrict__ rw,     // [E,H]
                           const float* __restrict__ bias,   // [E]
                           float* __restrict__ W)            // [T,E]
{
    const int lane = threadIdx.x & 31;
    const int wave = threadIdx.x >> 5;
    const int t    = blockIdx.x * 8 + wave;
    if (t >= T) return;

    float acc[E];
#pragma unroll
    for (int e = 0; e < E; ++e) acc[e] = 0.f;

    for (int h = lane; h < H; h += 32) {
        const float xv = x[(size_t)t * H + h];
#pragma unroll
        for (int e = 0; e < E; ++e) acc[e] += xv * rw[e * H + h];
    }
#pragma unroll
    for (int e = 0; e < E; ++e) {
        float v = acc[e];
#pragma unroll
        for (int off = 16; off > 0; off >>= 1) v += __shfl_down(v, off, 32);
        acc[e] = v;
    }
    if (lane == 0) {
        float p[E];
#pragma unroll
        for (int e = 0; e < E; ++e) p[e] = 1.f / (1.f + __expf(-(acc[e] + bias[e])));
        int i1 = 0; float s1 = p[0];
#pragma unroll
        for (int e = 1; e < E; ++e) if (p[e] > s1) { s1 = p[e]; i1 = e; }
        int i2 = -1; float s2 = -1.f;
#pragma unroll
        for (int e = 0; e < E; ++e) if (e != i1 && p[e] > s2) { s2 = p[e]; i2 = e; }
        const float inv = 1.f / (s1 + s2);
#pragma unroll
        for (int e = 0; e < E; ++e) W[(size_t)t * E + e] = 0.f;
        W[(size_t)t * E + i1] = s1 * inv;
        W[(size_t)t * E + i2] = s2 * inv;
    }
}

// B-tile loader into K-pair-packed LDS: element (k,n) at [(k>>1)*BSTRIDE + 2n + (k&1)]
// BT=false: B row-major [K,N]; BT=true: B is [N,K].
template <bool BT>
__device__ __forceinline__ void load_b_tile(float* __restrict__ Bs,
                                            const float* __restrict__ B,
                                            int k0, int n0, int ldb, int tid)
{
    if (!BT) {
        const int p  = tid >> 5;            // K pair 0..7
        const int nh = tid & 31;            // n half-index: covers n = 2*nh, 2*nh+1
        const int n  = 2 * nh;
        const v2f g0 = *(const v2f*)&B[(size_t)(k0 + 2 * p + 0) * ldb + n0 + n];
        const v2f g1 = *(const v2f*)&B[(size_t)(k0 + 2 * p + 1) * ldb + n0 + n];
        v2f lo; lo.x = g0.x; lo.y = g1.x;   // (k even, k odd) for column n
        v2f hi; hi.x = g0.y; hi.y = g1.y;   // for column n+1
        *(v2f*)&Bs[p * BSTRIDE + (n + 0) * 2] = lo;
        *(v2f*)&Bs[p * BSTRIDE + (n + 1) * 2] = hi;
    } else {
        const int nn = tid >> 2;            // 0..63
        const int kq = (tid & 3) * 4;       // 0,4,8,12
        const v4f bv = *(const v4f*)&B[(size_t)(n0 + nn) * ldb + k0 + kq];
        v2f lo; lo.x = bv.x; lo.y = bv.y;
        v2f hi; hi.x = bv.z; hi.y = bv.w;
        *(v2f*)&Bs[((kq >> 1) + 0) * BSTRIDE + nn * 2] = lo;
        *(v2f*)&Bs[((kq >> 1) + 1) * BSTRIDE + nn * 2] = hi;
    }
}

// ---------------------------------------------------------------------------
// Fused gate+up GEMM with SiLU(g)*u*w(token) epilogue into act[T,I].
// ---------------------------------------------------------------------------
template <bool BT, bool ROUTED>
__global__ void moe_gateup(const float* __restrict__ A,   // x [T,H]
                           const float* __restrict__ Bg,
                           const float* __restrict__ Bu,
                           float* __restrict__ Cact,      // act [T,I]
                           const float* __restrict__ W,   // [T,E] or null
                           int expert, int ldb)
{
    const int tid  = threadIdx.x;
    const int lane = tid & 31;
    const int wave = tid >> 5;                 // 0..7
    const int m0   = blockIdx.y * BM;          // token base
    const int n0   = blockIdx.x * BN;          // i base

    __shared__ float As[BM][LDA];
    __shared__ float Bgs[BP * BSTRIDE];
    __shared__ float Bus[BP * BSTRIDE];
    __shared__ float wTile[BM];
    __shared__ int   active;

    if (ROUTED) {
        if (tid == 0) active = 0;
        if (tid < BM) wTile[tid] = W[(size_t)(m0 + tid) * E + expert];
        __syncthreads();
        if (tid < BM && wTile[tid] != 0.f) atomicOr(&active, 1);
        __syncthreads();
        if (!active) {
            const v4f z = {0.f, 0.f, 0.f, 0.f};
            for (int r = tid; r < BM * (BN / 4); r += blockDim.x) {
                const int mm = r / (BN / 4);
                const int nq = (r % (BN / 4)) * 4;
                *(v4f*)&Cact[(size_t)(m0 + mm) * I + n0 + nq] = z;
            }
            return;
        }
    } else {
        if (tid < BM) wTile[tid] = 1.f;
    }

    // 8 waves tile the 64x64 block: wave -> (mIdx 0..3) x (two adjacent N tiles)
    const int mIdx = wave >> 1;
    const int nIdx = (wave & 1) * 2;

    v8f accg0 = {}, accg1 = {}, accu0 = {}, accu1 = {};

    const int mA = lane & 15;          // M within 16x16 tile (A operand)
    const int nA = lane & 15;          // N within 16x16 tile (B operand)
    const int kb = (lane >> 4) * 2;    // K-pair selector per ISA layout
    const int bcol0 = (nIdx * 16 + nA) * 2;
    const int bcol1 = ((nIdx + 1) * 16 + nA) * 2;

    for (int k0 = 0; k0 < H; k0 += BK) {
        // ---- A tile: 64 rows x 16 k, one float4 per thread
        {
            const int row = tid >> 2;
            const int kq  = (tid & 3) * 4;
            const v4f av = *(const v4f*)&A[(size_t)(m0 + row) * H + k0 + kq];
            *(v4f*)&As[row][kq] = av;
        }
        load_b_tile<BT>(Bgs, Bg, k0, n0, ldb, tid);
        load_b_tile<BT>(Bus, Bu, k0, n0, ldb, tid);
        __syncthreads();

        // prefetch next K tile (global_prefetch_b8)
        if (k0 + BK < H) {
            __builtin_prefetch(&A[(size_t)(m0 + (tid >> 2)) * H + k0 + BK], 0, 1);
            if (!BT)
                __builtin_prefetch(&Bg[(size_t)(k0 + BK + (tid >> 5)) * ldb + n0], 0, 1);
            else
                __builtin_prefetch(&Bg[(size_t)(n0 + (tid >> 2)) * ldb + k0 + BK], 0, 1);
        }

#pragma unroll
        for (int kk = 0; kk < BK; kk += 4) {
            const int pi = (kk + kb) >> 1;     // K-pair row (kk+kb is even)
            const v2f a   = *(const v2f*)&As[mIdx * 16 + mA][kk + kb];
            const v2f bg0 = *(const v2f*)&Bgs[pi * BSTRIDE + bcol0];
            const v2f bg1 = *(const v2f*)&Bgs[pi * BSTRIDE + bcol1];
            const v2f bu0 = *(const v2f*)&Bus[pi * BSTRIDE + bcol0];
            const v2f bu1 = *(const v2f*)&Bus[pi * BSTRIDE + bcol1];

            accg0 = __builtin_amdgcn_wmma_f32_16x16x4_f32(false, a, false, bg0,
                                                          (short)0, accg0, false, false);
            accg1 = __builtin_amdgcn_wmma_f32_16x16x4_f32(false, a, false, bg1,
                                                          (short)0, accg1, false, false);
            accu0 = __builtin_amdgcn_wmma_f32_16x16x4_f32(false, a, false, bu0,
                                                          (short)0, accu0, false, false);
            accu1 = __builtin_amdgcn_wmma_f32_16x16x4_f32(false, a, false, bu1,
                                                          (short)0, accu1, false, false);
        }
        __syncthreads();
    }

    // epilogue: silu(g)*u*w(token)
#pragma unroll
    for (int r = 0; r < 8; ++r) {
        const int mm = r + (lane >> 4) * 8;          // M within 16x16 C tile
        const int t  = m0 + mIdx * 16 + mm;
        const float wtok = wTile[mIdx * 16 + mm];
        {
            const float g = accg0[r], u = accu0[r];
            const float hv = (g / (1.f + __expf(-g))) * u * wtok;
            Cact[(size_t)t * I + n0 + nIdx * 16 + nA] = hv;
        }
        {
            const float g = accg1[r], u = accu1[r];
            const float hv = (g / (1.f + __expf(-g))) * u * wtok;
            Cact[(size_t)t * I + n0 + (nIdx + 1) * 16 + nA] = hv;
        }
    }
}

// ---------------------------------------------------------------------------
// Down projection: Out[t,h] (+)= sum_i act[t,i] * Bd[i,h]
// ---------------------------------------------------------------------------
template <bool BT, bool ACCUM, bool ROUTED>
__global__ void moe_down(const float* __restrict__ A,    // act [T,I]
                         const float* __restrict__ Bd,
                         float* __restrict__ Out,        // [T,H]
                         const float* __restrict__ W,
                         int expert, int ldb)
{
    const int tid  = threadIdx.x;
    const int lane = tid & 31;
    const int wave = tid >> 5;
    const int m0   = blockIdx.y * BM;
    const int n0   = blockIdx.x * BN;

    __shared__ float As[BM][LDA];
    __shared__ float Bs[BP * BSTRIDE];
    __shared__ int   active;

    if (ROUTED) {
        if (tid == 0) active = 0;
        __syncthreads();
        if (tid < BM && W[(size_t)(m0 + tid) * E + expert] != 0.f) atomicOr(&active, 1);
        __syncthreads();
        if (!active) return;   // expert contributes zero for this whole token tile
    }

    const int mIdx = wave >> 1;
    const int nIdx = (wave & 1) * 2;

    v8f acc0 = {}, acc1 = {};

    const int mA = lane & 15;
    const int nA = lane & 15;
    const int kb = (lane >> 4) * 2;
    const int bcol0 = (nIdx * 16 + nA) * 2;
    const int bcol1 = ((nIdx + 1) * 16 + nA) * 2;

    for (int k0 = 0; k0 < I; k0 += BK) {
        {
            const int row = tid >> 2;
            const int kq  = (tid & 3) * 4;
            const v4f av = *(const v4f*)&A[(size_t)(m0 + row) * I + k0 + kq];
            *(v4f*)&As[row][kq] = av;
        }
        load_b_tile<BT>(Bs, Bd, k0, n0, ldb, tid);
        __syncthreads();

        if (k0 + BK < I) {
            __builtin_prefetch(&A[(size_t)(m0 + (tid >> 2)) * I + k0 + BK], 0, 1);
            if (!BT)
                __builtin_prefetch(&Bd[(size_t)(k0 + BK + (tid >> 5)) * ldb + n0], 0, 1);
            else
                __builtin_prefetch(&Bd[(size_t)(n0 + (tid >> 2)) * ldb + k0 + BK], 0, 1);
        }

#pragma unroll
        for (int kk = 0; kk < BK; kk += 4) {
            const int pi = (kk + kb) >> 1;
            const v2f a  = *(const v2f*)&As[mIdx * 16 + mA][kk + kb];
            const v2f b0 = *(const v2f*)&Bs[pi * BSTRIDE + bcol0];
            const v2f b1 = *(const v2f*)&Bs[pi * BSTRIDE + bcol1];

            acc0 = __builtin_amdgcn_wmma_f32_16x16x4_f32(false, a, false, b0,
                                                         (short)0, acc0, false, false);
            acc1 = __builtin_amdgcn_wmma_f32_16x16x4_f32(false, a, false, b1,
                                                         (short)0, acc1, false, false);
        }
        __syncthreads();
    }

#pragma unroll
    for (int r = 0; r < 8; ++r) {
        const int mm = r + (lane >> 4) * 8;
        const int t  = m0 + mIdx * 16 + mm;
        {
            const size_t idx = (size_t)t * H + n0 + nIdx * 16 + nA;
            float v = acc0[r];
            if (ACCUM) v += Out[idx];
            Out[idx] = v;
        }
        {
            const size_t idx = (size_t)t * H + n0 + (nIdx + 1) * 16 + nA;
            float v = acc1[r];
            if (ACCUM) v += Out[idx];
            Out[idx] = v;
        }
    }
}

// ---------------------------------------------------------------------------
// Host-side orchestration (all launches on `stream`; graph-capture safe)
// ---------------------------------------------------------------------------
extern "C" void kernel_launch(void* const* d_in, const int* in_sizes, int n_in,
                              void* d_out, int out_size, void* d_ws, size_t ws_size,
                              hipStream_t stream)
{
    const float* x  = (const float*)d_in[0];   // [T,H]
    const float* sg = (const float*)d_in[1];   // [I,H]
    const float* su = (const float*)d_in[2];   // [I,H]
    const float* sd = (const float*)d_in[3];   // [H,I]
    const float* gw = (const float*)d_in[4];   // [E,H,I]
    const float* uw = (const float*)d_in[5];   // [E,H,I]
    const float* dw = (const float*)d_in[6];   // [E,I,H]
    const float* rw = (const float*)d_in[7];   // [E,H]
    const float* rb = (const float*)d_in[8];   // [E]
    float* out = (float*)d_out;                // [T,H]

    float* W   = (float*)d_ws;                 // [T,E]   (128 KB)
    float* act = W + (size_t)T * E;            // [T,I]   (32 MB)

    const dim3 blk(256);
    const dim3 gGU(I / BN, T / BM);
    const dim3 gDN(H / BN, T / BM);

    // 1) router -> dense per-token expert weights
    moe_router<<<T / 8, 256, 0, stream>>>(x, rw, rb, W);

    // 2) shared expert (weights stored [out,in] -> BT=true), writes out (beta=0)
    moe_gateup<true, false><<<gGU, blk, 0, stream>>>(x, sg, su, act, nullptr, 0, H);
    moe_down<true, false, false><<<gDN, blk, 0, stream>>>(act, sd, out, nullptr, 0, I);

    // 3) routed experts: weighted SwiGLU accumulated into out (beta=1)
    for (int e = 0; e < E; ++e) {
        moe_gateup<false, true><<<gGU, blk, 0, stream>>>(
            x, gw + (size_t)e * H * I, uw + (size_t)e * H * I, act, W, e, I);
        moe_down<false, true, true><<<gDN, blk, 0, stream>>>(
            act, dw + (size_t)e * I * H, out, W, e, H);
    }
}